// _PerFieldConv1DBranch_72060961292424
// MI455X (gfx1250) — compile-verified
//
#include <hip/hip_runtime.h>

typedef __attribute__((ext_vector_type(2))) float v2f;
typedef __attribute__((ext_vector_type(4))) float v4f;
typedef __attribute__((ext_vector_type(8))) float v8f;

#define NF    32
#define BATCH 128
#define TLEN  1024
#define HID   16
#define KW    5
#define DPF   4          // pooled dims per field
#define CT    512        // time chunk
#define XSP   524        // x stage pitch (chunk + conv1 halo + conv2 halo)
#define HTP   20         // h1T row pitch in floats (16 channels + 4 pad -> 16B-aligned rows,
                         //  conflict-free b64 reads & b128 writes)
#define HROWS 520        // h1T rows (516 needed)
#define XS_OFF   0
#define H_OFF    524                         // multiple of 4 -> rows stay 16B aligned
#define PART_OFF (H_OFF + HROWS * HTP)       // 10924
#define LDS_FLOATS (PART_OFF + HID * DPF * 2)  // 11052 floats ~= 43 KB

__global__ __launch_bounds__(128)
void branch_kernel(const float* __restrict__ x,
                   const float* __restrict__ w1, const float* __restrict__ b1,
                   const float* __restrict__ w2, const float* __restrict__ b2,
                   const float* __restrict__ pw, const float* __restrict__ pb,
                   float* __restrict__ out)
{
    __shared__ __align__(16) float lds[LDS_FLOATS];
    const int f    = blockIdx.x & (NF - 1);
    const int b    = blockIdx.x >> 5;
    const int tid  = threadIdx.x;
    const int lane = tid & 31;
    const int wv   = tid >> 5;        // wave id 0..3
    const int m    = lane & 15;       // A row / B-N column index
    const int half = lane >> 4;

    const float* xg  = x  + (size_t)b * TLEN * NF + f;     // x[b][t][f] = xg[t*NF]
    const float* w1f = w1 + f * HID * KW;
    const float* b1f = b1 + f * HID;
    const float* w2f = w2 + f * HID * HID * KW;
    const float* b2f = b2 + f * HID;

    // Per-lane A fragments for conv2: A_dk[m][ci] = w2[f][m][ci][dk].
    // K-step s: lane holds K = 4s + 2*half, +1 (ISA 16x4 fp32 A layout).
    float aw[40];
#pragma unroll
    for (int dk = 0; dk < KW; ++dk) {
#pragma unroll
        for (int s = 0; s < 4; ++s) {
            int ci = 4 * s + 2 * half;
            aw[(dk * 4 + s) * 2 + 0] = w2f[(m * HID + ci    ) * KW + dk];
            aw[(dk * 4 + s) * 2 + 1] = w2f[(m * HID + ci + 1) * KW + dk];
        }
    }
    // Bias fragment: D reg r belongs to row m = r + 8*half.
    float bb[8];
#pragma unroll
    for (int r = 0; r < 8; ++r) bb[r] = b2f[r + 8 * half];

#pragma unroll 1
    for (int ch = 0; ch < 2; ++ch) {
        const int tbase = ch * CT;

        // Stage x chunk with halo: xs[v] holds x at t = tbase + v - 6 (0 outside).
        for (int v = tid; v < XSP; v += 128) {
            int t = tbase + v - 6;
            lds[XS_OFF + v] = (t >= 0 && t < TLEN) ? xg[(size_t)t * NF] : 0.f;
        }
        __syncthreads();

        // conv1 + ReLU, stored t-major: h1T[u][ci] at H_OFF + u*HTP + ci,
        // u row holds t = tbase + u - 4 (forced 0 outside [0,T)).
        for (int u = tid; u < HROWS; u += 128) {
            int t = tbase + u - 4;
            bool valid = (t >= 0 && t < TLEN);
            float xv[KW];
#pragma unroll
            for (int j = 0; j < KW; ++j) xv[j] = lds[XS_OFF + u + j];
            float hloc[HID];
#pragma unroll
            for (int c = 0; c < HID; ++c) {
                float h = b1f[c];
#pragma unroll
                for (int j = 0; j < KW; ++j) h = fmaf(w1f[c * KW + j], xv[j], h);
                hloc[c] = valid ? fmaxf(h, 0.f) : 0.f;
            }
#pragma unroll
            for (int g = 0; g < 4; ++g) {
                v4f q = { hloc[4*g+0], hloc[4*g+1], hloc[4*g+2], hloc[4*g+3] };
                *(v4f*)(&lds[H_OFF + u * HTP + 4 * g]) = q;   // 16B-aligned, conflict-free
            }
        }
        __syncthreads();

        // conv2 via fp32 WMMA, fused bias+ReLU+pool accumulation.
        // Wave wv owns pool segment seg = 2*ch + (wv>>1), tiles j = (wv>>1)*16 + (wv&1)*8 + jj.
        float pacc[8];
#pragma unroll
        for (int r = 0; r < 8; ++r) pacc[r] = 0.f;

#pragma unroll 2
        for (int jj = 0; jj < 8; ++jj) {
            int j  = (wv >> 1) * 16 + (wv & 1) * 8 + jj;
            int ub = 16 * j + 2;   // local u for (n=0, dk=0): u = 16j + n + dk + 2
            v8f acc8 = {0.f, 0.f, 0.f, 0.f, 0.f, 0.f, 0.f, 0.f};
#pragma unroll
            for (int dk = 0; dk < KW; ++dk) {
                int rowbase = H_OFF + (ub + m + dk) * HTP + 2 * half;
#pragma unroll
                for (int s = 0; s < 4; ++s) {
                    v2f a;
                    a.x = aw[(dk * 4 + s) * 2 + 0];
                    a.y = aw[(dk * 4 + s) * 2 + 1];
                    // B pair (ci, ci+1) at same t: one aligned b64 load
                    v2f bv = *(const v2f*)(&lds[rowbase + 4 * s]);
                    acc8 = __builtin_amdgcn_wmma_f32_16x16x4_f32(
                               false, a, false, bv, (short)0, acc8, false, false);
                }
            }
#pragma unroll
            for (int r = 0; r < 8; ++r) pacc[r] += fmaxf(acc8[r] + bb[r], 0.f);
        }

        // Reduce across the 16 lanes (time columns) sharing each output row m.
        int seg = 2 * ch + (wv >> 1);
        int par = wv & 1;
#pragma unroll
        for (int r = 0; r < 8; ++r) {
            float v = pacc[r];
            v += __shfl_xor(v, 1, 32);
            v += __shfl_xor(v, 2, 32);
            v += __shfl_xor(v, 4, 32);
            v += __shfl_xor(v, 8, 32);
            if ((lane & 15) == 0)
                lds[PART_OFF + ((r + 8 * half) * DPF + seg) * 2 + par] = v;
        }
        __syncthreads();
    }

    // Final linear head: out[b][f*4+d'] = pb + pw[f][d'] . p_flat  (p H-major)
    if (tid < DPF) {
        const float* pwf = pw + ((size_t)f * DPF + tid) * (HID * DPF);
        float acc = pb[f * DPF + tid];
#pragma unroll
        for (int i = 0; i < HID * DPF; ++i) {
            float p = (lds[PART_OFF + i * 2] + lds[PART_OFF + i * 2 + 1]) * (1.f / 256.f);
            acc = fmaf(pwf[i], p, acc);
        }
        out[(size_t)b * (NF * DPF) + f * DPF + tid] = acc;
    }
}

extern "C" void kernel_launch(void* const* d_in, const int* in_sizes, int n_in,
                              void* d_out, int out_size, void* d_ws, size_t ws_size,
                              hipStream_t stream) {
    (void)in_sizes; (void)n_in; (void)d_ws; (void)ws_size; (void)out_size;
    const float* x  = (const float*)d_in[0];
    // d_in[1] = lengths (int32) -- unused by the reference
    const float* w1 = (const float*)d_in[2];
    const float* b1 = (const float*)d_in[3];
    const float* w2 = (const float*)d_in[4];
    const float* b2 = (const float*)d_in[5];
    const float* pw = (const float*)d_in[6];
    const float* pb = (const float*)d_in[7];
    float* out = (float*)d_out;

    dim3 grid(NF * BATCH);   // 4096 workgroups: one per (field, batch)
    dim3 block(128);         // 4 wave32s
    branch_kernel<<<grid, block, 0, stream>>>(x, w1, b1, w2, b2, pw, pb, out);
}